// BandedJointEncoder_90117003804693
// MI455X (gfx1250) — compile-verified
//
#include <hip/hip_runtime.h>
#include <cstdint>

// BandedJointEncoder: invert upper-bidiagonal precision matrices.
// B=4, T=512, Z=32, C=3*Z=96.
#define TT    512
#define ZC    32
#define CH    96
#define NTHR  128            // threads per block
#define COLS  (2 * NTHR)     // columns per block (2 per lane, float2 stores)

typedef uint32_t u32x4 __attribute__((ext_vector_type(4)));
typedef uint32_t u32x8 __attribute__((ext_vector_type(8)));

// cov kernel: block (c-tile, b*32+j). Lane owns columns c, c+1 (c = c0 + 2*tid).
// Sweep rows r = 0..511:  L[r,c] = w * (1/d_r)  (w==0 above diagonal);  w *= t_r.
__global__ __launch_bounds__(NTHR) void bje_cov_kernel(const float* __restrict__ mapped,
                                                       float* __restrict__ cov) {
  __shared__ float2 s_raw[TT];   // (d_raw, s_raw) pairs, filled by TDM
  __shared__ float2 s_proc[TT];  // (1/d, -s/d) pairs

  const int bz  = blockIdx.y;        // b*32 + j
  const int b   = bz >> 5;
  const int j   = bz & 31;
  const int c0  = blockIdx.x * COLS;
  const int tid = threadIdx.x;

  // ---- TDM: async-load the strided (512 x 2) fp32 tile into LDS (wave 0 only) ----
  if (tid < 32) {
    uint64_t ga  = (uint64_t)(uintptr_t)(mapped + ((size_t)b * TT * CH + (ZC + 2 * j)));
    uint32_t lds = (uint32_t)(uintptr_t)(&s_raw[0]);  // LDS aperture: low 32 bits = LDS offset

    u32x4 g0;                                          // D# group 0 (ISA 08 §8.3)
    g0[0] = 1u;                                        // count=1, user mode, no gather
    g0[1] = lds;                                       // lds_addr (bytes)
    g0[2] = (uint32_t)ga;                              // global_addr[31:0]
    g0[3] = (uint32_t)((ga >> 32) & 0x01FFFFFFu)       // global_addr[56:32]
          | (2u << 30);                                // type=2 ("image")

    u32x8 g1;                                          // D# group 1 (ISA 08 §8.4)
    g1[0] = (2u << 16);                                // workgroup_mask=0, data_size=2 (4B)
    g1[1] = (2u << 16);                                // tensor_dim0 = 2   (bits 79:48)
    g1[2] = ((uint32_t)TT << 16);                      // tensor_dim1 = 512 (bits 111:80)
    g1[3] = (2u << 16);                                // tile_dim0 = 2     (bits 127:112)
    g1[4] = (uint32_t)TT;                              // tile_dim1 = 512, tile_dim2 = 0
    g1[5] = (uint32_t)CH;                              // tensor_dim0_stride = 96 (data_size units)
    g1[6] = 0u;                                        // stride hi / dim1_stride lo = 0
    g1[7] = 0u;                                        // dim1_stride hi = 0

    // 2D tensor -> VADDR2/VADDR3 omitted (NULL)
    asm volatile("tensor_load_to_lds %0, %1" :: "s"(g0), "s"(g1) : "memory");
    __builtin_amdgcn_s_wait_tensorcnt(0);
  }
  __syncthreads();

  // ---- phase 2: per-row coefficients (IEEE divides, done once) ----
  for (int r = tid; r < TT; r += NTHR) {
    float2 p  = s_raw[r];
    float d   = p.x + 1.0f;
    float inv = 1.0f / d;
    float t   = (-p.y) / d;
    s_proc[r] = make_float2(inv, t);
  }
  __syncthreads();

  // ---- phase 3: row sweep, two columns per lane, coalesced b64 stores ----
  const int c = c0 + 2 * tid;
  float w0 = 0.0f, w1 = 0.0f;
  float2* rowp = (float2*)(cov + (size_t)bz * TT * TT + (size_t)c0) + tid;
#pragma unroll 4
  for (int r = 0; r < TT; ++r) {
    if (r == c)     w0 = 1.0f;
    if (r == c + 1) w1 = 1.0f;
    float2 p = s_proc[r];                 // ds_load_b64, same-address broadcast
    float v0 = w0 * p.x;                  // w==0 handles the upper triangle
    float v1 = w1 * p.x;
    v0 = __builtin_isfinite(v0) ? v0 : 0.0f;   // reference's isfinite masking
    v1 = __builtin_isfinite(v1) ? v1 : 0.0f;
    *rowp = make_float2(v0, v1);
    rowp += TT / 2;
    w0 *= p.y;
    w1 *= p.y;
  }
}

// mean kernel: mean[(b*32+j)*512 + t] = mapped[b, t, j]
__global__ __launch_bounds__(256) void bje_mean_kernel(const float* __restrict__ mapped,
                                                       float* __restrict__ mean) {
  int idx = blockIdx.x * 256 + threadIdx.x;
  int t  = idx & (TT - 1);
  int bz = idx >> 9;
  int b  = bz >> 5;
  int j  = bz & 31;
  mean[idx] = mapped[((size_t)b * TT + t) * CH + j];
}

extern "C" void kernel_launch(void* const* d_in, const int* in_sizes, int n_in,
                              void* d_out, int out_size, void* d_ws, size_t ws_size,
                              hipStream_t stream) {
  const float* mapped = (const float*)d_in[0];
  float* out = (float*)d_out;

  const int B = in_sizes[0] / (TT * CH);        // = 4
  const int n_mean = B * ZC * TT;               // 65536
  float* mean = out;
  float* cov  = out + n_mean;

  bje_mean_kernel<<<n_mean / 256, 256, 0, stream>>>(mapped, mean);

  dim3 grid(TT / COLS, B * ZC);                 // (2, 128) = 256 blocks
  bje_cov_kernel<<<grid, NTHR, 0, stream>>>(mapped, cov);
}